// Proj_wAttn_15908558864908
// MI455X (gfx1250) — compile-verified
//
#include <hip/hip_runtime.h>
#include <hip/hip_bf16.h>

typedef float v2f __attribute__((ext_vector_type(2)));
typedef float v8f __attribute__((ext_vector_type(8)));

#define H_STRIDE 136                     // floats between heads in LDS (pad: bank-conflict free)
#define N_STRIDE (16 * H_STRIDE + 4)     // 2180 floats between n-rows in LDS

__global__ __launch_bounds__(512, 1)
void proj_wattn_kernel(const float* __restrict__ x,
                       const float* __restrict__ attn,
                       const float* __restrict__ W,
                       const float* __restrict__ bias,
                       float* __restrict__ out)
{
    // x_proj tile: [16 n][16 h][128 e] fp32, padded.  ~136 KB (CDNA5: 320 KB/WGP)
    __shared__ float xproj[16 * N_STRIDE];

    const int b    = blockIdx.x >> 7;    // batch (8)
    const int nt   = blockIdx.x & 127;   // n-tile (128 tiles of 16 rows)
    const int wave = threadIdx.x >> 5;   // 16 waves per block (wave32)
    const int lane = threadIdx.x & 31;
    const int lrow = lane & 15;
    const int hiHalf = lane >> 4;        // 0 | 1
    const int khalf  = hiHalf << 1;      // 0 | 2  (A/B frag K offset for upper lanes)

    // ---------------- Phase 1: per-head GEMM  x[16,128] @ W[h][128,128] ----------------
    {
        const int h = wave;
        const float* xrow = x + (((size_t)(b * 2048 + nt * 16 + lrow)) << 11) + h * 128;
        const float* Wh   = W + (size_t)h * (128 * 128);
        const float* bh   = bias + h * 128;

        v8f acc[8];
#pragma unroll
        for (int t = 0; t < 8; ++t) {
            float bb = bh[16 * t + lrow];            // bias depends only on e (=N)
            v8f v = {bb, bb, bb, bb, bb, bb, bb, bb};
            acc[t] = v;
        }

        for (int s = 0; s < 32; ++s) {               // K = 128, 4 per WMMA
            const int k0 = 4 * s + khalf;
            v2f a = *(const v2f*)(xrow + k0);        // A frag: rows n=lrow, K=k0,k0+1
            const float* w0 = Wh + (size_t)k0 * 128; // W row k0 (e contiguous)
#pragma unroll
            for (int t = 0; t < 8; ++t) {            // 8 e-tiles of 16
                v2f bf;
                bf.x = w0[16 * t + lrow];            // K=k0,   N=e
                bf.y = w0[128 + 16 * t + lrow];      // K=k0+1, N=e
                acc[t] = __builtin_amdgcn_wmma_f32_16x16x4_f32(
                    false, a, false, bf, (short)0, acc[t], false, false);
            }
        }

        // Spill x_proj tile to LDS: [n][h][e]; strides padded so the two lane
        // halves (n=r vs n=r+8) land in disjoint bank sets.
#pragma unroll
        for (int t = 0; t < 8; ++t) {
#pragma unroll
            for (int r = 0; r < 8; ++r) {
                int n = r + (hiHalf << 3);
                xproj[n * N_STRIDE + h * H_STRIDE + 16 * t + lrow] = acc[t][r];
            }
        }
    }

    __syncthreads();

    // ---------------- Phase 2: head-mix  attn[16,16] @ x_proj[16,128] per row ----------
    {
        const int n = wave;                          // one n-row per wave
        const float* arow = attn + (((size_t)(b * 2048 + nt * 16 + n)) << 8); // [k][h] 16x16
        v8f acc[8];
#pragma unroll
        for (int t = 0; t < 8; ++t) {
            v8f z = {0.f, 0.f, 0.f, 0.f, 0.f, 0.f, 0.f, 0.f};
            acc[t] = z;
        }

#pragma unroll
        for (int s = 0; s < 4; ++s) {                // K = 16 heads, 4 per WMMA
            const int k0 = 4 * s + khalf;            // head index
            v2f a = *(const v2f*)(arow + lrow * 16 + k0);   // A: M=k rows, K=h
            const float* l0 = &xproj[n * N_STRIDE + k0 * H_STRIDE];
#pragma unroll
            for (int t = 0; t < 8; ++t) {
                v2f bf;
                bf.x = l0[16 * t + lrow];                   // K=k0,   N=e
                bf.y = l0[H_STRIDE + 16 * t + lrow];        // K=k0+1, N=e
                acc[t] = __builtin_amdgcn_wmma_f32_16x16x4_f32(
                    false, a, false, bf, (short)0, acc[t], false, false);
            }
        }

        float* orow = out + (((size_t)(b * 2048 + nt * 16 + n)) << 11); // [k*128 + e]
#pragma unroll
        for (int t = 0; t < 8; ++t) {
#pragma unroll
            for (int r = 0; r < 8; ++r) {
                int k = r + (hiHalf << 3);
                orow[k * 128 + 16 * t + lrow] = acc[t][r];
            }
        }
    }
}

extern "C" void kernel_launch(void* const* d_in, const int* in_sizes, int n_in,
                              void* d_out, int out_size, void* d_ws, size_t ws_size,
                              hipStream_t stream)
{
    const float* x    = (const float*)d_in[0];   // [8, 2048, 2048] f32
    const float* attn = (const float*)d_in[1];   // [8, 2048, 16, 16] f32
    const float* W    = (const float*)d_in[2];   // [16, 128, 128] f32
    const float* bias = (const float*)d_in[3];   // [16, 128] f32
    float* out = (float*)d_out;                  // [8, 2048, 2048] f32
    (void)in_sizes; (void)n_in; (void)out_size; (void)d_ws; (void)ws_size;

    dim3 grid(1024);   // 8 batches * 128 n-tiles
    dim3 block(512);   // 16 wave32 waves
    proj_wattn_kernel<<<grid, block, 0, stream>>>(x, attn, W, bias, out);
}